// GAE_Encoder_59330678227391
// MI455X (gfx1250) — compile-verified
//
#include <hip/hip_runtime.h>
#include <math.h>

// ---------------- problem constants (from reference) ----------------
#define NNODES  50000
#define NEDGES  800000
#define ETOT    (NEDGES + NNODES)   // with self loops: 850000
#define IN_DIM  128
#define HID     64
#define HEADS   4
#define OUT_DIM 64
#define F1      (HEADS * HID)       // 256
// order-preserving encoding of -inf (see enc_ord below): ~0xFF800000
#define ENC_NEG_INF 0x007FFFFFu

typedef __attribute__((ext_vector_type(2))) float v2f;
typedef __attribute__((ext_vector_type(8))) float v8f;

__device__ __forceinline__ float leakyf(float x, float s) { return x >= 0.f ? x : s * x; }

// monotone float<->uint mapping so unsigned atomicMax == float max
__device__ __forceinline__ unsigned enc_ord(float x) {
    unsigned u = __float_as_uint(x);
    return (u & 0x80000000u) ? ~u : (u | 0x80000000u);
}
__device__ __forceinline__ float dec_ord(unsigned e) {
    return (e & 0x80000000u) ? __uint_as_float(e & 0x7FFFFFFFu) : __uint_as_float(~e);
}

__device__ __forceinline__ void edge_sd(const int* __restrict__ ei, int t, int& s, int& d) {
    if (t < NEDGES) { s = ei[t]; d = ei[NEDGES + t]; }
    else            { s = t - NEDGES; d = s; }     // appended self loops
}

// ---------------- fp32 WMMA GEMM: C[nrows,fout] = A[nrows,fin] @ W[fin,fout] ----------------
// One wave -> one 16x64 strip (4 x 16x16 tiles sharing the A fragment).
// fin % 4 == 0, fout % 64 == 0, nrows % 16 == 0.
__global__ void gemm_wmma_f32_kernel(const float* __restrict__ A,
                                     const float* __restrict__ W,
                                     float* __restrict__ C,
                                     int nrows, int fin, int fout) {
    const int gwave = (blockIdx.x * blockDim.x + threadIdx.x) >> 5;
    const int lane  = threadIdx.x & 31;
    const int stripsPerRow = fout >> 6;
    const int totalStrips  = (nrows >> 4) * stripsPerRow;
    if (gwave >= totalStrips) return;              // wave-uniform exit: EXEC all-1s inside

    const int rowBlk = gwave / stripsPerRow;
    const int strip  = gwave % stripsPerRow;
    const int row0 = rowBlk << 4;
    const int col0 = strip << 6;
    const int half = lane >> 4;                    // K-pair selector (lanes 16-31 -> K+2)
    const int l    = lane & 15;                    // row (A) / col (B,C)

    v8f acc0 = {}, acc1 = {}, acc2 = {}, acc3 = {};
    const float* arow = A + (size_t)(row0 + l) * fin;
    for (int k = 0; k < fin; k += 4) {
        v2f a;
        a.x = arow[k + 2 * half + 0];              // A[m=l][k + 2*half + {0,1}]
        a.y = arow[k + 2 * half + 1];
        const float* w0 = W + (size_t)(k + 2 * half) * fout + col0 + l;
        const float* w1 = w0 + fout;
        v2f b0 = { w0[ 0], w1[ 0] };
        v2f b1 = { w0[16], w1[16] };
        v2f b2 = { w0[32], w1[32] };
        v2f b3 = { w0[48], w1[48] };
        acc0 = __builtin_amdgcn_wmma_f32_16x16x4_f32(false, a, false, b0, (short)0, acc0, false, false);
        acc1 = __builtin_amdgcn_wmma_f32_16x16x4_f32(false, a, false, b1, (short)0, acc1, false, false);
        acc2 = __builtin_amdgcn_wmma_f32_16x16x4_f32(false, a, false, b2, (short)0, acc2, false, false);
        acc3 = __builtin_amdgcn_wmma_f32_16x16x4_f32(false, a, false, b3, (short)0, acc3, false, false);
    }
    float* crow = C + (size_t)row0 * fout + col0;
#pragma unroll
    for (int r = 0; r < 8; ++r) {
        const int m = r + 8 * half;                // C/D layout: VGPR r holds rows r / r+8
        float* p = crow + (size_t)m * fout + l;
        p[ 0] = acc0[r];
        p[16] = acc1[r];
        p[32] = acc2[r];
        p[48] = acc3[r];
    }
}

// ---------------- attention logits per (node, head) ----------------
__global__ void alpha_kernel(const float* __restrict__ h,
                             const float* __restrict__ a_src, const float* __restrict__ a_dst,
                             float* __restrict__ as_, float* __restrict__ ad_,
                             int nn, int H, int Cc) {
    int idx = blockIdx.x * blockDim.x + threadIdx.x;
    if (idx >= nn * H) return;
    int n = idx / H, hh = idx - n * H;
    const float* hp = h + (size_t)n * H * Cc + hh * Cc;
    float s = 0.f, d = 0.f;
    for (int c = 0; c < Cc; ++c) {
        float v = hp[c];
        s += v * a_src[hh * Cc + c];
        d += v * a_dst[hh * Cc + c];
    }
    as_[idx] = s;
    ad_[idx] = d;
}

// ---------------- edge pass 1: e = leaky(as[src]+ad[dst]); segment max ----------------
__global__ void edge_max_kernel(const int* __restrict__ ei,
                                const float* __restrict__ as_, const float* __restrict__ ad_,
                                float* __restrict__ ebuf, unsigned* __restrict__ menc, int H) {
    int idx = blockIdx.x * blockDim.x + threadIdx.x;
    if (idx >= ETOT * H) return;
    int t = idx / H, hh = idx - t * H;
    int s, d; edge_sd(ei, t, s, d);
    float e = leakyf(as_[s * H + hh] + ad_[d * H + hh], 0.2f);
    ebuf[idx] = e;
    atomicMax(&menc[d * H + hh], enc_ord(e));
}

// ---------------- edge pass 2: ex = exp(e - m[dst]); segment sum ----------------
__global__ void edge_exp_kernel(const int* __restrict__ ei,
                                float* __restrict__ ebuf, const unsigned* __restrict__ menc,
                                float* __restrict__ den, int H) {
    int idx = blockIdx.x * blockDim.x + threadIdx.x;
    if (idx >= ETOT * H) return;
    int t = idx / H, hh = idx - t * H;
    int s, d; edge_sd(ei, t, s, d);
    (void)s;
    float m  = dec_ord(menc[d * H + hh]);
    float ex = expf(ebuf[idx] - m);
    ebuf[idx] = ex;
    atomicAdd(&den[d * H + hh], ex);
}

// ---------------- edge pass 3: out[dst] += alpha * h[src]  (one wave per edge) ----------------
__global__ void scatter_kernel(const int* __restrict__ ei,
                               const float* __restrict__ h,
                               const float* __restrict__ ebuf,
                               const float* __restrict__ den,
                               float* __restrict__ out, int H) {
    const int gwave = (blockIdx.x * blockDim.x + threadIdx.x) >> 5;
    const int lane  = threadIdx.x & 31;
    if (gwave >= ETOT) return;
    int s, d; edge_sd(ei, gwave, s, d);
    const int F = H << 6;                              // Cc == 64 in both layers
    const float* hs = h + (size_t)s * F;
    float* od = out + (size_t)d * F;
    float alph[HEADS];
#pragma unroll
    for (int hh = 0; hh < HEADS; ++hh)
        if (hh < H) alph[hh] = ebuf[(size_t)gwave * H + hh] / (den[d * H + hh] + 1e-16f);
    for (int c = lane; c < F; c += 32)
        atomicAdd(&od[c], alph[c >> 6] * hs[c]);
}

// ---------------- epilogue: out = leaky(in + bias, slope)  (slope=1 -> identity) ----------------
__global__ void bias_act_kernel(const float* __restrict__ in, const float* __restrict__ b,
                                float* __restrict__ out, int total, int F, float slope) {
    int i = blockIdx.x * blockDim.x + threadIdx.x;
    if (i >= total) return;
    out[i] = leakyf(in[i] + b[i % F], slope);
}

__global__ void fill_u32_kernel(unsigned* __restrict__ p, unsigned v, int n) {
    int i = blockIdx.x * blockDim.x + threadIdx.x;
    if (i < n) p[i] = v;
}

extern "C" void kernel_launch(void* const* d_in, const int* in_sizes, int n_in,
                              void* d_out, int out_size, void* d_ws, size_t ws_size,
                              hipStream_t stream) {
    const float* x   = (const float*)d_in[0];
    const int*   ei  = (const int*)  d_in[1];   // [2, E] row-major: src then dst
    const float* W1  = (const float*)d_in[2];
    const float* as1 = (const float*)d_in[3];
    const float* ad1 = (const float*)d_in[4];
    const float* b1  = (const float*)d_in[5];
    const float* W2  = (const float*)d_in[6];
    const float* as2 = (const float*)d_in[7];
    const float* ad2 = (const float*)d_in[8];
    const float* b2  = (const float*)d_in[9];
    float* out = (float*)d_out;

    // workspace layout (floats); layer-2 aliases layer-1 buffers that are dead by then
    float* ws   = (float*)d_ws;
    float* h1   = ws;                                  // N*256
    float* h2   = h1   + (size_t)NNODES * F1;          // N*256 (out1, then leaky in place)
    float* ebuf = h2   + (size_t)NNODES * F1;          // Et*4
    float* asb  = ebuf + (size_t)ETOT * HEADS;         // N*4
    float* adb  = asb  + (size_t)NNODES * HEADS;       // N*4
    unsigned* menc = (unsigned*)(adb + (size_t)NNODES * HEADS);  // N*4
    float* den  = (float*)(menc + (size_t)NNODES * HEADS);       // N*4
    float* g2   = h1;                                  // layer-2 features (N*64), reuses h1
    float* out2 = h1 + (size_t)NNODES * OUT_DIM;       // layer-2 aggregate (N*64)

    // ================= layer 1 (4 heads x 64 ch) =================
    {
        int strips = (NNODES / 16) * (F1 / 64);        // 12500 waves
        gemm_wmma_f32_kernel<<<(strips + 3) / 4, 128, 0, stream>>>(x, W1, h1, NNODES, IN_DIM, F1);
    }
    alpha_kernel<<<(NNODES * HEADS + 255) / 256, 256, 0, stream>>>(h1, as1, ad1, asb, adb,
                                                                   NNODES, HEADS, HID);
    hipMemsetAsync(den, 0, sizeof(float) * (size_t)NNODES * HEADS, stream);
    fill_u32_kernel<<<(NNODES * HEADS + 255) / 256, 256, 0, stream>>>(menc, ENC_NEG_INF,
                                                                      NNODES * HEADS);
    hipMemsetAsync(h2, 0, sizeof(float) * (size_t)NNODES * F1, stream);
    edge_max_kernel<<<(ETOT * HEADS + 255) / 256, 256, 0, stream>>>(ei, asb, adb, ebuf, menc, HEADS);
    edge_exp_kernel<<<(ETOT * HEADS + 255) / 256, 256, 0, stream>>>(ei, ebuf, menc, den, HEADS);
    scatter_kernel<<<(ETOT + 7) / 8, 256, 0, stream>>>(ei, h1, ebuf, den, h2, HEADS);
    bias_act_kernel<<<(NNODES * F1 + 255) / 256, 256, 0, stream>>>(h2, b1, h2,
                                                                   NNODES * F1, F1, 0.01f);

    // ================= layer 2 (1 head x 64 ch) =================
    {
        int strips = (NNODES / 16) * (OUT_DIM / 64);   // 3125 waves
        gemm_wmma_f32_kernel<<<(strips + 3) / 4, 128, 0, stream>>>(h2, W2, g2, NNODES, F1, OUT_DIM);
    }
    alpha_kernel<<<(NNODES + 255) / 256, 256, 0, stream>>>(g2, as2, ad2, asb, adb,
                                                           NNODES, 1, OUT_DIM);
    hipMemsetAsync(den, 0, sizeof(float) * (size_t)NNODES, stream);
    fill_u32_kernel<<<(NNODES + 255) / 256, 256, 0, stream>>>(menc, ENC_NEG_INF, NNODES);
    hipMemsetAsync(out2, 0, sizeof(float) * (size_t)NNODES * OUT_DIM, stream);
    edge_max_kernel<<<(ETOT + 255) / 256, 256, 0, stream>>>(ei, asb, adb, ebuf, menc, 1);
    edge_exp_kernel<<<(ETOT + 255) / 256, 256, 0, stream>>>(ei, ebuf, menc, den, 1);
    scatter_kernel<<<(ETOT + 7) / 8, 256, 0, stream>>>(ei, g2, ebuf, den, out2, 1);
    bias_act_kernel<<<(NNODES * OUT_DIM + 255) / 256, 256, 0, stream>>>(out2, b2, out,
                                                                        NNODES * OUT_DIM,
                                                                        OUT_DIM, 1.0f);
}